// LSTM1_11304353923792
// MI455X (gfx1250) — compile-verified
//
#include <hip/hip_runtime.h>

// MI455X / gfx1250, wave32. Fused LSTM(13->7, T=512) + FF head.
// Latency-bound serial recurrence (T=512): per-step critical path is
// h-WMMAs -> LDS -> activations -> LDS -> next h-WMMAs. The x-projection
// (8 WMMAs + 7 global loads) for step t+1 is software-pipelined into the
// shadow of step t's cell update. Activations are branchless single-TRANS
// sequences (v_tanh_f32 if exposed, else v_exp_f32/v_rcp_f32), and the cell
// update is made branch-free via clamped indices + LDS pointer selects so
// the whole timestep loop has no EXEC save/restore.

typedef float v2f __attribute__((ext_vector_type(2)));
typedef float v8f __attribute__((ext_vector_type(8)));

#define BATCH 4096
#define TSTEP 512
#define IDIM  13
#define HDIM  7
#define GDIM  28   // 4*H
#define FFDIM 512

#define LOG2E 1.4426950408889634f

// ---- activation tiers: v_tanh_f32 > exp2/rcp TRANS ops > libm ----
#if __has_builtin(__builtin_amdgcn_tanhf)
#define FAST_TANH(v) __builtin_amdgcn_tanhf(v)
#define TANH_TIER 2
#elif __has_builtin(__builtin_amdgcn_tanh_f32)
#define FAST_TANH(v) __builtin_amdgcn_tanh_f32(v)
#define TANH_TIER 2
#elif __has_builtin(__builtin_amdgcn_exp2f) && __has_builtin(__builtin_amdgcn_rcpf)
#define TANH_TIER 1
static __device__ __forceinline__ float FAST_TANH(float v) {
    // tanh(x) = 1 - 2/(e^{2x}+1); saturates correctly via rcp(inf)=0, rcp of
    // (0+1)=1. Branchless: 1 mul + v_exp_f32 + add + v_rcp_f32 + fma.
    float e2 = __builtin_amdgcn_exp2f(v * (2.0f * LOG2E));
    return fmaf(-2.0f, __builtin_amdgcn_rcpf(e2 + 1.0f), 1.0f);
}
#else
#define FAST_TANH(v) tanhf(v)
#define TANH_TIER 0
#endif

static __device__ __forceinline__ float sigm(float v) {
#if TANH_TIER == 2
    // sigmoid(x) = 0.5 + 0.5*tanh(x/2): one TRANS op, no divide, no branches
    return fmaf(0.5f, FAST_TANH(0.5f * v), 0.5f);
#elif TANH_TIER == 1
    return __builtin_amdgcn_rcpf(1.0f + __builtin_amdgcn_exp2f(v * -LOG2E));
#else
    return 1.0f / (1.0f + __expf(-v));
#endif
}

__global__ __launch_bounds__(32) void lstm_fused_kernel(
    const float* __restrict__ x,     const float* __restrict__ w_ih,
    const float* __restrict__ w_hh,  const float* __restrict__ b_ih,
    const float* __restrict__ b_hh,  const float* __restrict__ fc1_w,
    const float* __restrict__ fc1_b, const float* __restrict__ fc_w,
    const float* __restrict__ fc_b,  float* __restrict__ out)
{
    __shared__ float gates[16][32];  // 16 rows x 28 gate cols (padded to 32)
    __shared__ float hbuf[16][8];    // h state, col 7 = zero pad (K-pad for WMMA)
    __shared__ float hdump[16];      // trash slot for the phantom 4th unit (half 1)
    __shared__ float red[16];        // head partial reduction

    const int l    = threadIdx.x;    // 0..31
    const int half = l >> 4;         // lane half selects K sub-pair in A/B layout
    const int nl   = l & 15;
    const int b0   = blockIdx.x * 16;

    // ---- Build WMMA B operands in registers (held for the whole T loop) ----
    // B layout (KxN, 32-bit): lane holds N = l%16; VGPR v holds K = v + 2*(l>=16).
    v2f Bih[2][4];   // [n-tile][k-chunk], K = kc*4 + 2*half + {0,1}, zero padded
    v2f Bhh[2][2];
#pragma unroll
    for (int kc = 0; kc < 4; ++kc) {
        int k0 = kc * 4 + 2 * half;
#pragma unroll
        for (int nt = 0; nt < 2; ++nt) {
            int n  = nt * 16 + nl;
            int nn = (n < GDIM) ? n : 0;
            float m0 = (n < GDIM && k0     < IDIM) ? 1.0f : 0.0f;
            float m1 = (n < GDIM && k0 + 1 < IDIM) ? 1.0f : 0.0f;
            int ka = (k0     < IDIM) ? k0     : 0;
            int kb = (k0 + 1 < IDIM) ? k0 + 1 : 0;
            Bih[nt][kc][0] = w_ih[nn * IDIM + ka] * m0;
            Bih[nt][kc][1] = w_ih[nn * IDIM + kb] * m1;
        }
    }
#pragma unroll
    for (int kc = 0; kc < 2; ++kc) {
        int k0 = kc * 4 + 2 * half;
#pragma unroll
        for (int nt = 0; nt < 2; ++nt) {
            int n  = nt * 16 + nl;
            int nn = (n < GDIM) ? n : 0;
            float m0 = (n < GDIM && k0     < HDIM) ? 1.0f : 0.0f;
            float m1 = (n < GDIM && k0 + 1 < HDIM) ? 1.0f : 0.0f;
            int ka = (k0     < HDIM) ? k0     : 0;
            int kb = (k0 + 1 < HDIM) ? k0 + 1 : 0;
            Bhh[nt][kc][0] = w_hh[nn * HDIM + ka] * m0;
            Bhh[nt][kc][1] = w_hh[nn * HDIM + kb] * m1;
        }
    }

    // Bias folded into the WMMA accumulator init: C(M,N) = bias[N], N = l%16 (+16).
    const float bias0 = b_ih[nl] + b_hh[nl];
    const float bias1 = (16 + nl < GDIM) ? (b_ih[16 + nl] + b_hh[16 + nl]) : 0.0f;

    if (half == 0) {
#pragma unroll
        for (int k = 0; k < 8; ++k) hbuf[nl][k] = 0.0f;  // h0 = 0 (+K pad)
    }
    float c[4] = {0.0f, 0.0f, 0.0f, 0.0f};  // cell state in registers
    __syncthreads();

    // A-operand row for x loads: lane holds batch row b0 + l%16.
    const float* xrow = x + (size_t)(b0 + nl) * (TSTEP * IDIM);

    // x-projection for one timestep: bias + x_t @ w_ih^T (independent of h).
    auto x_project = [&](int t, v8f& xa0, v8f& xa1) {
        const float* xt = xrow + t * IDIM;
#pragma unroll
        for (int i = 0; i < 8; ++i) { xa0[i] = bias0; xa1[i] = bias1; }
#pragma unroll
        for (int kc = 0; kc < 4; ++kc) {
            int k0 = kc * 4 + 2 * half;
            v2f a;
            // clamp+mask (no divergent branch: EXEC must stay all-1s for WMMA)
            int ka = (k0     < IDIM) ? k0     : 0;
            int kb = (k0 + 1 < IDIM) ? k0 + 1 : 0;
            a[0] = xt[ka] * ((k0     < IDIM) ? 1.0f : 0.0f);
            a[1] = xt[kb] * ((k0 + 1 < IDIM) ? 1.0f : 0.0f);
            xa0 = __builtin_amdgcn_wmma_f32_16x16x4_f32(false, a, false, Bih[0][kc],
                                                        (short)0, xa0, false, false);
            xa1 = __builtin_amdgcn_wmma_f32_16x16x4_f32(false, a, false, Bih[1][kc],
                                                        (short)0, xa1, false, false);
        }
    };

    v8f xacc0, xacc1;
    x_project(0, xacc0, xacc1);          // pipeline prologue

    for (int t = 0; t < TSTEP; ++t) {
        v8f acc0 = xacc0, acc1 = xacc1;

        // gates += h_{t-1} @ w_hh^T  (K = 7 padded to 8; 2 chunks of K=4)
#pragma unroll
        for (int kc = 0; kc < 2; ++kc) {
            int k0 = kc * 4 + 2 * half;
            v2f a;
            a[0] = hbuf[nl][k0];
            a[1] = hbuf[nl][k0 + 1];
            acc0 = __builtin_amdgcn_wmma_f32_16x16x4_f32(false, a, false, Bhh[0][kc],
                                                         (short)0, acc0, false, false);
            acc1 = __builtin_amdgcn_wmma_f32_16x16x4_f32(false, a, false, Bhh[1][kc],
                                                         (short)0, acc1, false, false);
        }

        // C/D layout: VGPR v of lane l = element (M = v + 8*(l>=16), N = l%16 [+16]).
#pragma unroll
        for (int v = 0; v < 8; ++v) {
            int m = v + 8 * half;
            gates[m][nl]      = acc0[v];
            gates[m][nl + 16] = acc1[v];
        }
        __syncthreads();   // single-wave workgroup: cheap, orders LDS

        // Software pipeline: start step t+1's x-projection now; it has no
        // dependence on h_t, so its loads + 8 WMMAs overlap the cell update
        // below (WMMA co-executes with VALU). Clamp index on last iteration
        // (uniform select) to avoid reading past the end of x.
        int tn = (t + 1 < TSTEP) ? (t + 1) : 0;
        x_project(tn, xacc0, xacc1);

        // Cell update: lane pair (r, r+16) owns batch row r; j split 0..3 / 4..6.
        // Branch-free: all lanes run 4 unit-updates; the phantom unit (half=1,
        // jj=3 -> j=7) clamps its gate reads to j=6 and steers its h store to a
        // trash LDS slot via pointer select (v_cndmask on address, EXEC untouched).
        {
            const int row = nl;
            const int jb  = half * 4;
#pragma unroll
            for (int jj = 0; jj < 4; ++jj) {
                int j  = jb + jj;                       // 0..3 / 4..7
                int jc = (j < HDIM) ? j : (HDIM - 1);   // clamp phantom reads
                float ig = sigm(gates[row][jc]);
                float fg = sigm(gates[row][jc + 7]);
                float gg = FAST_TANH(gates[row][jc + 14]);
                float og = sigm(gates[row][jc + 21]);
                c[jj] = fg * c[jj] + ig * gg;
                float hv = og * FAST_TANH(c[jj]);
                float* dst = (j < HDIM) ? &hbuf[row][j] : &hdump[row];
                *dst = hv;
            }
        }
        __syncthreads();
    }

    // ---- Head: out = relu(relu(hn) @ fc1^T + b1) @ fc^T + b ----
    float hr[HDIM];
#pragma unroll
    for (int j = 0; j < HDIM; ++j) {
        float v = hbuf[nl][j];
        hr[j] = v > 0.0f ? v : 0.0f;
    }
    float acc = 0.0f;
    const int ffb = half * (FFDIM / 2);
    for (int ff = ffb; ff < ffb + FFDIM / 2; ++ff) {
        float s = fc1_b[ff];
#pragma unroll
        for (int j = 0; j < HDIM; ++j) s += fc1_w[ff * HDIM + j] * hr[j];
        s = s > 0.0f ? s : 0.0f;
        acc += fc_w[ff] * s;
    }
    if (half == 1) red[nl] = acc;
    __syncthreads();
    if (half == 0) out[b0 + nl] = acc + red[nl] + fc_b[0];
}

extern "C" void kernel_launch(void* const* d_in, const int* in_sizes, int n_in,
                              void* d_out, int out_size, void* d_ws, size_t ws_size,
                              hipStream_t stream) {
    (void)in_sizes; (void)n_in; (void)out_size; (void)d_ws; (void)ws_size;
    const float* x     = (const float*)d_in[0];
    const float* w_ih  = (const float*)d_in[1];
    const float* w_hh  = (const float*)d_in[2];
    const float* b_ih  = (const float*)d_in[3];
    const float* b_hh  = (const float*)d_in[4];
    const float* fc1_w = (const float*)d_in[5];
    const float* fc1_b = (const float*)d_in[6];
    const float* fc_w  = (const float*)d_in[7];
    const float* fc_b  = (const float*)d_in[8];

    lstm_fused_kernel<<<dim3(BATCH / 16), dim3(32), 0, stream>>>(
        x, w_ih, w_hh, b_ih, b_hh, fc1_w, fc1_b, fc_w, fc_b, (float*)d_out);
}